// KalmanFilter_2302102470889
// MI455X (gfx1250) — compile-verified
//
#include <hip/hip_runtime.h>
#include <hip/hip_bf16.h>

// Kalman filter forward for MI455X (gfx1250).
// One wave32 = one batch element. All 12x12 / 4x12 matrix products are padded
// into a single 16x16 WMMA tile and run on v_wmma_f32_16x16x4_f32 (full f32
// precision -- required for 128 recursive covariance updates). LDS stages the
// per-wave covariance and intermediates; transposed operands are free via LDS
// indexed loads.

#define LDW 18  // row stride (floats) for 16x16 LDS tiles: conflict-free, 8B aligned

typedef float v2f __attribute__((ext_vector_type(2)));
typedef float v8f __attribute__((ext_vector_type(8)));

__device__ __forceinline__ v8f wmma4(v2f a, v2f b, v8f c) {
  // 8 args: (neg_a, A, neg_b, B, c_mod, C, reuse_a, reuse_b)
  return __builtin_amdgcn_wmma_f32_16x16x4_f32(false, a, false, b, (short)0, c,
                                               false, false);
}

// A fragment (16x4, k-block kb): lanes 0-15 hold M=0..15 / K=4kb+{0,1};
// lanes 16-31 hold K=4kb+{2,3}.
__device__ __forceinline__ v2f loadA(const float* mat, int ld, int kb, int lane,
                                     bool transpose) {
  int row = lane & 15;
  int k0 = 4 * kb + 2 * (lane >> 4);
  v2f a;
  if (!transpose) {
    a.x = mat[row * ld + k0];
    a.y = mat[row * ld + k0 + 1];
  } else {
    a.x = mat[k0 * ld + row];
    a.y = mat[(k0 + 1) * ld + row];
  }
  return a;
}

// B fragment (4x16, k-block kb): lanes 0-15 N=0..15 / K=4kb+{0,1}; lanes 16-31 K=4kb+{2,3}.
__device__ __forceinline__ v2f loadB(const float* mat, int ld, int kb, int lane,
                                     bool transpose) {
  int col = lane & 15;
  int k0 = 4 * kb + 2 * (lane >> 4);
  v2f b;
  if (!transpose) {
    b.x = mat[k0 * ld + col];
    b.y = mat[(k0 + 1) * ld + col];
  } else {
    b.x = mat[col * ld + k0];
    b.y = mat[col * ld + k0 + 1];
  }
  return b;
}

// C/D layout: VGPR r, lanes 0-15 -> (row=r, col=lane); lanes 16-31 -> (row=r+8, col=lane-16)
__device__ __forceinline__ void storeC(float* mat, int ld, int lane, v8f c) {
  int hi = lane >> 4;
  int col = lane & 15;
#pragma unroll
  for (int r = 0; r < 8; ++r) mat[(r + 8 * hi) * ld + col] = c[r];
}

struct PerWave {
  float P[16 * LDW];   // covariance (prior/filtered), 16x16 padded
  float W[16 * LDW];   // workspace (HP / PHt / T1 / T2)
  float E[16 * LDW];   // I - K H
  float KG[16 * 4];    // Kn = -K (negated Kalman gain), 16x4
  float KR[16 * 4];    // Kn * R
  float Sinv[4 * 16];  // negated 4x4 inverse, zero-padded to 4x16 (B operand)
  float Sm[16];        // innovation covariance 4x4
  float m[16];         // one-step-ahead mean
  float mf[16];        // filtered mean
  float resid[4];
};

__global__ __launch_bounds__(128) void kalman_wmma_kernel(
    const float* __restrict__ inp, const float* __restrict__ Fg,
    const float* __restrict__ Hg, const float* __restrict__ Qg,
    const float* __restrict__ Rg, const float* __restrict__ m0g,
    const float* __restrict__ P0g, float* __restrict__ out, int B, int T) {
  __shared__ float Fs[16 * LDW];  // F, 16x16 padded
  __shared__ float Hs[16 * 16];   // H, rows 4-15 zero
  __shared__ float Rs[4 * 16];    // R as B operand, cols 4-15 zero
  __shared__ PerWave pw[4];

  const int lane = threadIdx.x & 31;
  const int wave = threadIdx.x >> 5;
  const int b = blockIdx.x * 4 + wave;

  // ---- block-cooperative constant staging ----
  for (int idx = threadIdx.x; idx < 16 * LDW; idx += blockDim.x) {
    int i = idx / LDW, j = idx % LDW;
    Fs[idx] = (i < 12 && j < 12) ? Fg[i * 12 + j] : 0.0f;
  }
  for (int idx = threadIdx.x; idx < 256; idx += blockDim.x) {
    int i = idx >> 4, j = idx & 15;
    Hs[idx] = (i < 4 && j < 12) ? Hg[i * 12 + j] : 0.0f;
  }
  for (int idx = threadIdx.x; idx < 64; idx += blockDim.x) {
    int i = idx >> 4, j = idx & 15;
    Rs[idx] = (j < 4) ? Rg[i * 4 + j] : 0.0f;
  }
  __syncthreads();

  PerWave& w = pw[wave];

  // ---- per-wave state init ----
  for (int idx = lane; idx < 16 * LDW; idx += 32) w.P[idx] = 0.0f;
  for (int idx = lane; idx < 64; idx += 32) w.Sinv[idx] = 0.0f;
  for (int idx = lane; idx < 144; idx += 32)
    w.P[(idx / 12) * LDW + (idx % 12)] = P0g[(size_t)b * 144 + idx];
  if (lane < 16) w.mf[lane] = (lane < 12) ? m0g[b * 12 + lane] : 0.0f;
  __builtin_amdgcn_wave_barrier();

  // per-lane constant C-layout fragments: Q, R, I
  const int chi = lane >> 4;
  const int ccol = lane & 15;
  v8f cQ, cR, cI;
#pragma unroll
  for (int r = 0; r < 8; ++r) {
    int row = r + 8 * chi;
    cQ[r] = (row < 12 && ccol < 12) ? Qg[row * 12 + ccol] : 0.0f;
    cR[r] = (row < 4 && ccol < 4) ? Rg[row * 4 + ccol] : 0.0f;
    cI[r] = (row == ccol) ? 1.0f : 0.0f;
  }
  const v8f zero = {};

  for (int t = 0; t < T; ++t) {
    // ================= predict: P = F*Pf*F^T + Q ; m = F*mf =================
    v8f acc = zero;
#pragma unroll
    for (int kb = 0; kb < 3; ++kb)
      acc = wmma4(loadA(Fs, LDW, kb, lane, false),
                  loadB(w.P, LDW, kb, lane, false), acc);  // T2 = F*Pf
    storeC(w.W, LDW, lane, acc);
    acc = cQ;
#pragma unroll
    for (int kb = 0; kb < 3; ++kb)
      acc = wmma4(loadA(w.W, LDW, kb, lane, false),
                  loadB(Fs, LDW, kb, lane, true), acc);  // P = T2*F^T + Q
    storeC(w.P, LDW, lane, acc);

    if (lane < 12) {
      float s = 0.0f;
#pragma unroll
      for (int j = 0; j < 12; ++j) s += Fs[lane * LDW + j] * w.mf[j];
      w.m[lane] = s;
    }
    __builtin_amdgcn_wave_barrier();

    // ================= output y = H*m ; resid =================
    if (lane < 4) {
      float y = 0.0f;
#pragma unroll
      for (int s2 = 0; s2 < 12; ++s2) y += Hs[lane * 16 + s2] * w.m[s2];
      out[(size_t)t * B * 4 + (size_t)b * 4 + lane] = y;
      w.resid[lane] = inp[(size_t)b * T * 4 + (size_t)t * 4 + lane] - y;
    }
    __builtin_amdgcn_wave_barrier();

    // ================= update =================
    // HP = H*P
    acc = zero;
#pragma unroll
    for (int kb = 0; kb < 3; ++kb)
      acc = wmma4(loadA(Hs, 16, kb, lane, false),
                  loadB(w.P, LDW, kb, lane, false), acc);
    storeC(w.W, LDW, lane, acc);
    // Sm = HP*H^T + R
    acc = cR;
#pragma unroll
    for (int kb = 0; kb < 3; ++kb)
      acc = wmma4(loadA(w.W, LDW, kb, lane, false),
                  loadB(Hs, 16, kb, lane, true), acc);
    if (lane < 4) {
#pragma unroll
      for (int r = 0; r < 4; ++r) w.Sm[r * 4 + lane] = acc[r];
    }
    __builtin_amdgcn_wave_barrier();

    // 4x4 inverse (adjugate), stored NEGATED into Sinv (so Kn = -K below)
    {
      float a00 = w.Sm[0], a01 = w.Sm[1], a02 = w.Sm[2], a03 = w.Sm[3];
      float a10 = w.Sm[4], a11 = w.Sm[5], a12 = w.Sm[6], a13 = w.Sm[7];
      float a20 = w.Sm[8], a21 = w.Sm[9], a22 = w.Sm[10], a23 = w.Sm[11];
      float a30 = w.Sm[12], a31 = w.Sm[13], a32 = w.Sm[14], a33 = w.Sm[15];
      float b0 = a22 * a33 - a23 * a32;
      float b1 = a21 * a33 - a23 * a31;
      float b2 = a21 * a32 - a22 * a31;
      float b3 = a20 * a33 - a23 * a30;
      float b4 = a20 * a32 - a22 * a30;
      float b5 = a20 * a31 - a21 * a30;
      float det = a00 * (a11 * b0 - a12 * b1 + a13 * b2) -
                  a01 * (a10 * b0 - a12 * b3 + a13 * b4) +
                  a02 * (a10 * b1 - a11 * b3 + a13 * b5) -
                  a03 * (a10 * b2 - a11 * b4 + a12 * b5);
      float rdet = 1.0f / det;
      if (lane < 16) {
        int i = lane >> 2, j = lane & 3;
        int r0 = (j > 0) ? 0 : 1, r1 = (j > 1) ? 1 : 2, r2 = (j > 2) ? 2 : 3;
        int c0 = (i > 0) ? 0 : 1, c1 = (i > 1) ? 1 : 2, c2 = (i > 2) ? 2 : 3;
        float m00 = w.Sm[r0 * 4 + c0], m01 = w.Sm[r0 * 4 + c1],
              m02 = w.Sm[r0 * 4 + c2];
        float m10 = w.Sm[r1 * 4 + c0], m11 = w.Sm[r1 * 4 + c1],
              m12 = w.Sm[r1 * 4 + c2];
        float m20 = w.Sm[r2 * 4 + c0], m21 = w.Sm[r2 * 4 + c1],
              m22 = w.Sm[r2 * 4 + c2];
        float Md = m00 * (m11 * m22 - m12 * m21) -
                   m01 * (m10 * m22 - m12 * m20) +
                   m02 * (m10 * m21 - m11 * m20);
        float sgn = ((i + j) & 1) ? 1.0f : -1.0f;  // includes global negation
        w.Sinv[i * 16 + j] = sgn * Md * rdet;
      }
    }
    __builtin_amdgcn_wave_barrier();

    // PHt = P*H^T
    acc = zero;
#pragma unroll
    for (int kb = 0; kb < 3; ++kb)
      acc = wmma4(loadA(w.P, LDW, kb, lane, false),
                  loadB(Hs, 16, kb, lane, true), acc);
    storeC(w.W, LDW, lane, acc);
    // Kn = PHt * (-Sinv) = -K
    acc = wmma4(loadA(w.W, LDW, 0, lane, false),
                loadB(w.Sinv, 16, 0, lane, false), zero);
    if ((lane & 15) < 4) {
#pragma unroll
      for (int r = 0; r < 8; ++r)
        w.KG[(r + 8 * chi) * 4 + (lane & 3)] = acc[r];
    }
    __builtin_amdgcn_wave_barrier();

    // mf = m + K*resid = m - Kn*resid
    if (lane < 16) {
      float s = 0.0f;
      if (lane < 12) {
        s = w.m[lane];
#pragma unroll
        for (int j = 0; j < 4; ++j) s -= w.KG[lane * 4 + j] * w.resid[j];
      }
      w.mf[lane] = s;
    }
    __builtin_amdgcn_wave_barrier();

    // E = I + Kn*H = I - K*H
    acc = wmma4(loadA(w.KG, 4, 0, lane, false), loadB(Hs, 16, 0, lane, false),
                cI);
    storeC(w.E, LDW, lane, acc);
    // T1 = E*P
    acc = zero;
#pragma unroll
    for (int kb = 0; kb < 3; ++kb)
      acc = wmma4(loadA(w.E, LDW, kb, lane, false),
                  loadB(w.P, LDW, kb, lane, false), acc);
    storeC(w.W, LDW, lane, acc);
    // Kr = Kn*R
    acc = wmma4(loadA(w.KG, 4, 0, lane, false), loadB(Rs, 16, 0, lane, false),
                zero);
    if ((lane & 15) < 4) {
#pragma unroll
      for (int r = 0; r < 8; ++r)
        w.KR[(r + 8 * chi) * 4 + (lane & 3)] = acc[r];
    }
    __builtin_amdgcn_wave_barrier();

    // Pf = T1*E^T + Kr*Kn^T   (Kn R Kn^T == K R K^T)
    acc = zero;
#pragma unroll
    for (int kb = 0; kb < 3; ++kb)
      acc = wmma4(loadA(w.W, LDW, kb, lane, false),
                  loadB(w.E, LDW, kb, lane, true), acc);
    acc = wmma4(loadA(w.KR, 4, 0, lane, false), loadB(w.KG, 4, 0, lane, true),
                acc);
    storeC(w.P, LDW, lane, acc);
    __builtin_amdgcn_wave_barrier();
  }
}

extern "C" void kernel_launch(void* const* d_in, const int* in_sizes, int n_in,
                              void* d_out, int out_size, void* d_ws,
                              size_t ws_size, hipStream_t stream) {
  const float* inp = (const float*)d_in[0];
  const float* F = (const float*)d_in[1];
  const float* H = (const float*)d_in[2];
  const float* Q = (const float*)d_in[3];
  const float* R = (const float*)d_in[4];
  const float* m0 = (const float*)d_in[5];
  const float* P0 = (const float*)d_in[6];
  float* out = (float*)d_out;

  constexpr int B = 16384, T = 128;
  dim3 grid(B / 4), block(128);
  hipLaunchKernelGGL(kalman_wmma_kernel, grid, block, 0, stream, inp, F, H, Q,
                     R, m0, P0, out, B, T);
}